// FourierChannelAttention1d_81054622810780
// MI455X (gfx1250) — compile-verified
//
#include <hip/hip_runtime.h>
#include <hip/hip_bf16.h>

// ---------------------------------------------------------------------------
// FourierChannelAttention1d on gfx1250 (MI455X)
//   B=32, C=256, T=4096, ST=2049, LATENT=256, OUTC=64, BANDS=11
// Pipeline (all on `stream`):
//   1) k_fft_fwd : per (b,c) row: pack -> 2048pt Stockham FFT (LDS) -> rfft
//                  untangle (ortho) -> spectrum to ws; fused 2049->64 mag
//                  interp + log1p + alpha -> h0 (incl. grid channel)
//   2) k_conv3   : WMMA f32 16x16x4 GEMM conv (k=3, pad=1) + bias + PReLU (x2)
//                  kw-outer decomposition, LDS-staged activations; mask-free
//                  main K loop + single masked tail (uniform scalar branch)
//   3) k_pool    : adaptive max pool 64 -> 11
//   4) k_squeeze : 1x1 conv 256->64 + PReLU
//   5) k_latent  : FC 704->256 + PReLU
//   6) k_spat    : FC 256->2816 + sigmoid/0.4+0.5  (11-band gains)
//   7) k_chan    : FC 256->256 + sigmoid+0.1 clamp (channel gains)
//   8) k_fft_inv : per (b,c): band-gain interp 11->2049, scale spectrum,
//                  repack -> inverse 2048pt FFT -> irfft, channel gain, store
// ---------------------------------------------------------------------------

#define NB   32
#define NC   256
#define NT   4096
#define NM   2048      // complex FFT length
#define NST  2049
#define NLAT 256
#define NOC  64
#define NBAND 11
#define PI_F 3.14159265358979323846f

typedef __attribute__((ext_vector_type(2))) float v2f;
typedef __attribute__((ext_vector_type(8))) float v8f;

// ------------------------------- forward FFT -------------------------------
__global__ __launch_bounds__(256) void k_fft_fwd(
    const float* __restrict__ x,      // [B][C][T]
    const float* __restrict__ alpha,  // [C]
    float2* __restrict__ spec,        // [B*C][2049]
    float*  __restrict__ h0)          // [B][257][64]
{
    __shared__ float2 bufA[NM];
    __shared__ float2 bufB[NM];
    __shared__ float  magL[NST];

    const int bc  = blockIdx.x;
    const int b   = bc >> 8, c = bc & 255;
    const int tid = threadIdx.x;
    const float2* xp2 = (const float2*)(x + (size_t)bc * NT);

    // pack real 4096 -> complex 2048 (b64 loads)
    for (int m = tid; m < NM; m += 256)
        bufA[m] = xp2[m];
    __syncthreads();

    // Stockham radix-2, forward (sign = -1), 11 stages
    float2* src = bufA; float2* dst = bufB;
    for (int p = 1; p < NM; p <<= 1) {
        for (int t = tid; t < NM/2; t += 256) {
            int k = t & (p - 1);
            float ang = -PI_F * (float)k / (float)p;  // e^{-2pi i k/(2p)}
            float wr, wi; __sincosf(ang, &wi, &wr);
            float2 a  = src[t];
            float2 bb = src[t + NM/2];
            float br = wr*bb.x - wi*bb.y;
            float bi = wr*bb.y + wi*bb.x;
            int d0 = 2*t - k;
            dst[d0]     = make_float2(a.x + br, a.y + bi);
            dst[d0 + p] = make_float2(a.x - br, a.y - bi);
        }
        __syncthreads();
        float2* tmp = src; src = dst; dst = tmp;
    }

    // untangle to rfft(4096) with ortho norm; fused magnitude
    const float2* Z = src;
    const float inv_sqrtN = 1.0f / 64.0f;
    float2* sp = spec + (size_t)bc * NST;
    for (int k = tid; k <= NM; k += 256) {
        int km = k & (NM - 1);
        int kr = (NM - k) & (NM - 1);
        float2 zk = Z[km];
        float2 zr = Z[kr];
        float Er = 0.5f*(zk.x + zr.x), Ei = 0.5f*(zk.y - zr.y);
        float dr = zk.x - zr.x, di = zk.y + zr.y;   // zk - conj(zr)
        float Or = 0.5f*di, Oi = -0.5f*dr;          // -0.5i * (dr + i di)
        float ang = -PI_F * (float)k / (float)NM;   // e^{-2pi i k / 4096}
        float wr, wi; __sincosf(ang, &wi, &wr);
        float Xr = (Er + wr*Or - wi*Oi) * inv_sqrtN;
        float Xi = (Ei + wr*Oi + wi*Or) * inv_sqrtN;
        sp[k] = make_float2(Xr, Xi);
        magL[k] = sqrtf(Xr*Xr + Xi*Xi);
    }
    __syncthreads();

    // interp 2049 -> 64, log1p + alpha*mag  (F.interpolate linear, align=false)
    if (tid < 64) {
        float s = (tid + 0.5f) * ((float)NST / 64.0f) - 0.5f;
        s = fminf(fmaxf(s, 0.0f), (float)(NST - 1));
        int i0 = (int)floorf(s);
        int i1 = min(i0 + 1, NST - 1);
        float w = s - (float)i0;
        float mv = magL[i0]*(1.0f - w) + magL[i1]*w;
        h0[((size_t)b*257 + c)*64 + tid] = log1pf(mv) + alpha[c]*mv;
        if (c == 0)  // grid channel = linspace(0,1,64)
            h0[((size_t)b*257 + 256)*64 + tid] = (float)tid * (1.0f/63.0f);
    }
}

// ------------------- k=3 pad=1 conv as WMMA f32 GEMM -----------------------
// Y[b,m,n] = PReLU( sum_{kw<3} sum_{ic<Cin} W[m, ic*3+kw] * X[b, ic, n+kw-1] )
// One block per (n_tile, batch): activations staged once in LDS (18 cols,
// edge zeros materialized), 8 waves x 2 m-tiles each, 2 WMMAs share one B.
// Main K loop has no masks at all; the Cin%4 tail is one masked iteration
// behind a uniform scalar branch (EXEC stays all-ones for WMMA).
__global__ __launch_bounds__(256) void k_conv3_wmma(
    const float* __restrict__ W,      // [256][Cin*3]  (kw innermost)
    const float* __restrict__ bias,   // [256]
    const float* __restrict__ slope,  // [256]
    const float* __restrict__ X,      // [B][Cin][64]
    float* __restrict__ Y,            // [B][256][64]
    int Cin)
{
    __shared__ float Xs[257][18];     // [ic][t0-1 .. t0+16]

    const int nt = blockIdx.x;        // 0..3  (64 positions / 16)
    const int b  = blockIdx.y;        // batch
    const int tid  = threadIdx.x;
    const int wave = tid >> 5;        // 0..7
    const int lane = tid & 31;
    const int half = lane >> 4;       // K sub-pair select
    const int lml  = lane & 15;

    const int t0 = nt * 16;
    const float* Xb = X + (size_t)b * Cin * 64;

    // stage activation tile with halo; zeros at t<0 / t>=64
    for (int idx = tid; idx < Cin * 18; idx += 256) {
        int ic = idx / 18, j = idx - 18 * ic;
        int t = t0 + j - 1;
        Xs[ic][j] = (t >= 0 && t < 64) ? Xb[ic * 64 + t] : 0.0f;
    }
    __syncthreads();

    const int m0 = wave * 16;         // this wave's two output row-tiles
    const int m1 = (wave + 8) * 16;
    const int Kt = Cin * 3;
    const float* W0 = W + (size_t)(m0 + lml) * Kt;
    const float* W1 = W + (size_t)(m1 + lml) * Kt;

    v8f acc0 = {0.f,0.f,0.f,0.f,0.f,0.f,0.f,0.f};
    v8f acc1 = {0.f,0.f,0.f,0.f,0.f,0.f,0.f,0.f};

    const int CinFloor = Cin & ~3;    // mask-free portion
    for (int kw = 0; kw < 3; ++kw) {
        // ---- main body: no bounds masks ----
        for (int kk = 0; kk < CinFloor; kk += 4) {
            int k0 = kk + 2 * half;
            int k1 = k0 + 1;
            v2f bv, a0, a1;
            bv.x = Xs[k0][lml + kw];
            bv.y = Xs[k1][lml + kw];
            a0.x = W0[k0 * 3 + kw];
            a0.y = W0[k1 * 3 + kw];
            a1.x = W1[k0 * 3 + kw];
            a1.y = W1[k1 * 3 + kw];
            acc0 = __builtin_amdgcn_wmma_f32_16x16x4_f32(
                       false, a0, false, bv, (short)0, acc0, false, false);
            acc1 = __builtin_amdgcn_wmma_f32_16x16x4_f32(
                       false, a1, false, bv, (short)0, acc1, false, false);
        }
        // ---- tail: one masked iteration (uniform branch; EXEC unchanged) ----
        if (CinFloor < Cin) {
            int k0 = CinFloor + 2 * half;
            int k1 = k0 + 1;
            float f0 = (k0 < Cin) ? 1.0f : 0.0f;
            float f1 = (k1 < Cin) ? 1.0f : 0.0f;
            int c0 = min(k0, Cin - 1);
            int c1 = min(k1, Cin - 1);
            v2f bv, a0, a1;
            bv.x = Xs[c0][lml + kw] * f0;
            bv.y = Xs[c1][lml + kw] * f1;
            a0.x = W0[c0 * 3 + kw] * f0;
            a0.y = W0[c1 * 3 + kw] * f1;
            a1.x = W1[c0 * 3 + kw] * f0;
            a1.y = W1[c1 * 3 + kw] * f1;
            acc0 = __builtin_amdgcn_wmma_f32_16x16x4_f32(
                       false, a0, false, bv, (short)0, acc0, false, false);
            acc1 = __builtin_amdgcn_wmma_f32_16x16x4_f32(
                       false, a1, false, bv, (short)0, acc1, false, false);
        }
    }

    // D layout: VGPR j, lane L -> row = j + 8*(L/16), col = L%16
    const int n = t0 + lml;
    #pragma unroll
    for (int j = 0; j < 8; ++j) {
        int mm0 = m0 + j + 8 * half;
        float v = acc0[j] + bias[mm0];
        v = (v >= 0.0f) ? v : slope[mm0] * v;
        Y[((size_t)b * 256 + mm0) * 64 + n] = v;
        int mm1 = m1 + j + 8 * half;
        float u = acc1[j] + bias[mm1];
        u = (u >= 0.0f) ? u : slope[mm1] * u;
        Y[((size_t)b * 256 + mm1) * 64 + n] = u;
    }
}

// --------------------------- small scalar stages ---------------------------
__global__ void k_pool(const float* __restrict__ h2, float* __restrict__ pooled)
{
    int idx = blockIdx.x*blockDim.x + threadIdx.x;
    if (idx >= NB*NC*NBAND) return;
    int band = idx % NBAND; int bc = idx / NBAND;
    int s = (band*64) / NBAND;
    int e = ((band + 1)*64 + NBAND - 1) / NBAND;
    const float* p = h2 + (size_t)bc * 64;
    float mv = p[s];
    for (int i = s + 1; i < e; ++i) mv = fmaxf(mv, p[i]);
    pooled[idx] = mv;
}

__global__ void k_squeeze(const float* __restrict__ wc, const float* __restrict__ bcv,
                          const float* __restrict__ ac, const float* __restrict__ pooled,
                          float* __restrict__ e)
{
    int idx = blockIdx.x*blockDim.x + threadIdx.x;
    if (idx >= NB*NOC*NBAND) return;
    int band = idx % NBAND; int t = idx / NBAND;
    int oc = t % NOC; int b = t / NOC;
    const float* w = wc + (size_t)oc * NC;
    const float* p = pooled + (size_t)b * NC * NBAND + band;
    float s = 0.0f;
    for (int ic = 0; ic < NC; ++ic) s += w[ic] * p[ic*NBAND];
    s += bcv[oc];
    e[idx] = (s >= 0.0f) ? s : ac[oc]*s;
}

__global__ void k_latent(const float* __restrict__ we, const float* __restrict__ be,
                         const float* __restrict__ ae, const float* __restrict__ e,
                         float* __restrict__ latent)
{
    int idx = blockIdx.x*blockDim.x + threadIdx.x;
    if (idx >= NB*NLAT) return;
    int l = idx % NLAT; int b = idx / NLAT;
    const float* ev = e + (size_t)b * (NOC*NBAND);
    const float* w  = we + (size_t)l * (NOC*NBAND);
    float s = 0.0f;
    for (int j = 0; j < NOC*NBAND; ++j) s += ev[j]*w[j];
    s += be[l];
    latent[idx] = (s >= 0.0f) ? s : ae[l]*s;
}

__global__ void k_spat(const float* __restrict__ wf1, const float* __restrict__ bf1,
                       const float* __restrict__ latent, float* __restrict__ spat11)
{
    int idx = blockIdx.x*blockDim.x + threadIdx.x;
    if (idx >= NB*NC*NBAND) return;
    int r = idx % (NC*NBAND); int b = idx / (NC*NBAND);
    const float* lat = latent + (size_t)b * NLAT;
    const float* w   = wf1 + (size_t)r * NLAT;
    float s = 0.0f;
    for (int l = 0; l < NLAT; ++l) s += lat[l]*w[l];
    s += bf1[r];
    float sg = 1.0f / (1.0f + __expf(-s));
    spat11[idx] = sg * 2.5f + 0.5f;          // sigmoid/0.4 + 0.5
}

__global__ void k_chan(const float* __restrict__ wf2, const float* __restrict__ bf2,
                       const float* __restrict__ latent, float* __restrict__ chan)
{
    int idx = blockIdx.x*blockDim.x + threadIdx.x;
    if (idx >= NB*NC) return;
    int c = idx % NC; int b = idx / NC;
    const float* lat = latent + (size_t)b * NLAT;
    const float* w   = wf2 + (size_t)c * NLAT;
    float s = 0.0f;
    for (int l = 0; l < NLAT; ++l) s += lat[l]*w[l];
    s += bf2[c];
    float sg = 1.0f / (1.0f + __expf(-s)) + 0.1f;
    chan[idx] = fmaxf(sg, 1e-6f);
}

// ------------------------------- inverse FFT -------------------------------
__global__ __launch_bounds__(256) void k_fft_inv(
    const float2* __restrict__ spec,   // [B*C][2049]
    const float*  __restrict__ spat11, // [B*C][11]
    const float*  __restrict__ chan,   // [B*C]
    float* __restrict__ out)           // [B][C][T]
{
    __shared__ float2 Ybuf[NST];
    __shared__ float2 bufA[NM];
    __shared__ float2 bufB[NM];
    __shared__ float  sp[NBAND];

    const int bc  = blockIdx.x;
    const int tid = threadIdx.x;
    if (tid < NBAND) sp[tid] = spat11[(size_t)bc*NBAND + tid];
    __syncthreads();

    // scale spectrum by band gain (interp 11 -> 2049); polar(mag*s, phi) == s*X
    const float2* S = spec + (size_t)bc * NST;
    for (int k = tid; k <= NM; k += 256) {
        float s = (k + 0.5f) * ((float)NBAND / (float)NST) - 0.5f;
        s = fminf(fmaxf(s, 0.0f), (float)(NBAND - 1));
        int i0 = (int)floorf(s);
        int i1 = min(i0 + 1, NBAND - 1);
        float w = s - (float)i0;
        float sv = fmaxf(sp[i0]*(1.0f - w) + sp[i1]*w, 1e-6f);
        float2 v = S[k];
        Ybuf[k] = make_float2(v.x*sv, v.y*sv);
    }
    __syncthreads();

    // repack: Z[k] = E + iO,  E=0.5(Y[k]+Y[k+M]),  O=0.5 e^{+2pi i k/N}(Y[k]-Y[k+M])
    for (int k = tid; k < NM; k += 256) {
        float2 yk = Ybuf[k];
        float2 ym;
        if (k == 0) ym = Ybuf[NM];
        else { float2 t = Ybuf[NM - k]; ym = make_float2(t.x, -t.y); }
        float Er = 0.5f*(yk.x + ym.x), Ei = 0.5f*(yk.y + ym.y);
        float Dr = 0.5f*(yk.x - ym.x), Di = 0.5f*(yk.y - ym.y);
        float ang = PI_F * (float)k / (float)NM;
        float wr, wi; __sincosf(ang, &wi, &wr);
        float Or = wr*Dr - wi*Di;
        float Oi = wr*Di + wi*Dr;
        bufA[k] = make_float2(Er - Oi, Ei + Or);
    }
    __syncthreads();

    // Stockham radix-2, inverse (sign = +1)
    float2* src = bufA; float2* dst = bufB;
    for (int p = 1; p < NM; p <<= 1) {
        for (int t = tid; t < NM/2; t += 256) {
            int k = t & (p - 1);
            float ang = PI_F * (float)k / (float)p;
            float wr, wi; __sincosf(ang, &wi, &wr);
            float2 a  = src[t];
            float2 bb = src[t + NM/2];
            float br = wr*bb.x - wi*bb.y;
            float bi = wr*bb.y + wi*bb.x;
            int d0 = 2*t - k;
            dst[d0]     = make_float2(a.x + br, a.y + bi);
            dst[d0 + p] = make_float2(a.x - br, a.y - bi);
        }
        __syncthreads();
        float2* tmp = src; src = dst; dst = tmp;
    }

    // z_out = (sqrt(N)/M) * sum = 1/32 ; apply channel gain, unpack (b64 stores)
    const float cf = chan[bc] * (1.0f/32.0f);
    float2* op2 = (float2*)(out + (size_t)bc * NT);
    for (int m = tid; m < NM; m += 256) {
        float2 z = src[m];
        op2[m] = make_float2(z.x * cf, z.y * cf);
    }
}

// ------------------------------- launcher ----------------------------------
extern "C" void kernel_launch(void* const* d_in, const int* in_sizes, int n_in,
                              void* d_out, int out_size, void* d_ws, size_t ws_size,
                              hipStream_t stream)
{
    const float* x     = (const float*)d_in[0];
    const float* alpha = (const float*)d_in[1];
    const float* w1    = (const float*)d_in[2];
    const float* b1    = (const float*)d_in[3];
    const float* a1    = (const float*)d_in[4];
    const float* w2    = (const float*)d_in[5];
    const float* b2    = (const float*)d_in[6];
    const float* a2    = (const float*)d_in[7];
    const float* wc    = (const float*)d_in[8];
    const float* bcv   = (const float*)d_in[9];
    const float* ac    = (const float*)d_in[10];
    const float* we    = (const float*)d_in[11];
    const float* be    = (const float*)d_in[12];
    const float* ae    = (const float*)d_in[13];
    const float* wf1   = (const float*)d_in[14];
    const float* bf1   = (const float*)d_in[15];
    const float* wf2   = (const float*)d_in[16];
    const float* bf2   = (const float*)d_in[17];
    float* out = (float*)d_out;

    // workspace layout (floats)
    float* ws = (float*)d_ws;
    size_t off = 0;
    float2* spec  = (float2*)(ws + off); off += (size_t)NB*NC*NST*2;  // ~134 MB
    float* h0     = ws + off;            off += (size_t)NB*257*64;
    float* h1     = ws + off;            off += (size_t)NB*NC*64;
    float* h2     = ws + off;            off += (size_t)NB*NC*64;
    float* pooled = ws + off;            off += (size_t)NB*NC*NBAND;
    float* ebuf   = ws + off;            off += (size_t)NB*NOC*NBAND;
    float* latent = ws + off;            off += (size_t)NB*NLAT;
    float* spat11 = ws + off;            off += (size_t)NB*NC*NBAND;
    float* chan   = ws + off;            off += (size_t)NB*NC;

    // 1) forward rfft + fused 64-bin magnitude features
    k_fft_fwd<<<NB*NC, 256, 0, stream>>>(x, alpha, spec, h0);

    // 2) conv1 (Cin=257) and conv2 (Cin=256), WMMA GEMM + bias + PReLU
    dim3 cgrid(4, NB);
    k_conv3_wmma<<<cgrid, 256, 0, stream>>>(w1, b1, a1, h0, h1, 257);
    k_conv3_wmma<<<cgrid, 256, 0, stream>>>(w2, b2, a2, h1, h2, 256);

    // 3) adaptive max pool 64 -> 11
    k_pool<<<(NB*NC*NBAND + 255)/256, 256, 0, stream>>>(h2, pooled);

    // 4) 1x1 squeeze conv 256 -> 64 + PReLU
    k_squeeze<<<(NB*NOC*NBAND + 255)/256, 256, 0, stream>>>(wc, bcv, ac, pooled, ebuf);

    // 5) latent FC 704 -> 256 + PReLU
    k_latent<<<(NB*NLAT + 255)/256, 256, 0, stream>>>(we, be, ae, ebuf, latent);

    // 6) band gains FC 256 -> 2816, sigmoid/0.4 + 0.5
    k_spat<<<(NB*NC*NBAND + 255)/256, 256, 0, stream>>>(wf1, bf1, latent, spat11);

    // 7) channel gains FC 256 -> 256, sigmoid + 0.1, clamp
    k_chan<<<(NB*NC + 255)/256, 256, 0, stream>>>(wf2, bf2, latent, chan);

    // 8) scale spectrum, inverse rfft, channel gain
    k_fft_inv<<<NB*NC, 256, 0, stream>>>(spec, spat11, chan, out);
}